// AIRGCNIILayer_86294482911942
// MI455X (gfx1250) — compile-verified
//
#include <hip/hip_runtime.h>
#include <hip/hip_bf16.h>
#include <math.h>

#define NNODES 40000
#define NEDGES 640000
#define DIM    128
#define BETA   0.1f

typedef __attribute__((ext_vector_type(2))) float v2f;
typedef __attribute__((ext_vector_type(8))) float v8f;

// ---------------------------------------------------------------------------
// Kernel 1: zero the degree + agg regions of the workspace
// ---------------------------------------------------------------------------
__global__ void k_zero(float4* p, long n4) {
    long i = (long)blockIdx.x * blockDim.x + threadIdx.x;
    long stride = (long)gridDim.x * blockDim.x;
    float4 z = make_float4(0.f, 0.f, 0.f, 0.f);
    for (; i < n4; i += stride) p[i] = z;
}

// ---------------------------------------------------------------------------
// Kernel 2: degree count via float atomics (deg[dst[e]] += 1)
// ---------------------------------------------------------------------------
__global__ void k_degree(const int* __restrict__ dst, float* __restrict__ deg, int E) {
    int e = blockIdx.x * blockDim.x + threadIdx.x;
    if (e < E) atomicAdd(&deg[dst[e]], 1.0f);
}

// ---------------------------------------------------------------------------
// Kernel 3: deg -> norm = clip(deg,1)^(-1/2), in place
// ---------------------------------------------------------------------------
__global__ void k_norm(float* __restrict__ deg, int n) {
    int i = blockIdx.x * blockDim.x + threadIdx.x;
    if (i < n) deg[i] = rsqrtf(fmaxf(deg[i], 1.0f));
}

// ---------------------------------------------------------------------------
// Kernel 4: W_eff = BETA*Wl + (1-BETA)*I   (folds the residual into GEMM 3)
// ---------------------------------------------------------------------------
__global__ void k_weff(const float* __restrict__ Wl, float* __restrict__ Weff) {
    int i = blockIdx.x * blockDim.x + threadIdx.x;   // 0 .. 128*128-1
    int j = i >> 7, k = i & 127;
    float v = BETA * Wl[i];
    if (j == k) v += (1.0f - BETA);
    Weff[i] = v;
}

// ---------------------------------------------------------------------------
// Kernel 5: per-edge scatter. One wave32 per edge; lane handles 4 floats.
//           agg[dst] += features[src] * norm[src]
// ---------------------------------------------------------------------------
__global__ void k_scatter(const float* __restrict__ feat,
                          const int* __restrict__ src, const int* __restrict__ dst,
                          const float* __restrict__ norm, float* __restrict__ agg, int E) {
    int lane = threadIdx.x & 31;
    int wave = threadIdx.x >> 5;
    int e = blockIdx.x * 8 + wave;
    if (e >= E) return;
    int s = src[e], d = dst[e];
    float ns = norm[s];
    const float4 f = *(const float4*)(feat + (long)s * DIM + lane * 4);
    float* ap = agg + (long)d * DIM + lane * 4;
    atomicAdd(ap + 0, f.x * ns);
    atomicAdd(ap + 1, f.y * ns);
    atomicAdd(ap + 2, f.z * ns);
    atomicAdd(ap + 3, f.w * ns);
}

// ---------------------------------------------------------------------------
// Kernel 6: H = agg * norm[row]  (in place; agg becomes H)
// ---------------------------------------------------------------------------
__global__ void k_scale_rows(float4* __restrict__ agg4, const float* __restrict__ norm) {
    long i = (long)blockIdx.x * blockDim.x + threadIdx.x;  // N * (DIM/4) threads
    int row = (int)(i >> 5);                                // DIM/4 == 32 float4 per row
    float nv = norm[row];
    float4 v = agg4[i];
    v.x *= nv; v.y *= nv; v.z *= nv; v.w *= nv;
    agg4[i] = v;
}

// ---------------------------------------------------------------------------
// WMMA f32 16x16 tile GEMM:  C += A[m0:m0+16, 0:128] * W^T[:, n0:n0+16]
//   A row-major, lda = 128.  W row-major 128x128 used as B^T (B[k][j]=W[j][k]).
// A-frag (16x4 f32, 2 VGPRs): lanes 0-15: M=lane, K=k,k+1 ; lanes 16-31: M=lane-16, K=k+2,k+3
// B-frag (4x16 f32, 2 VGPRs): lanes 0-15: K=k,k+1 @ N=lane ; lanes 16-31: K=k+2,k+3 @ N=lane-16
// ---------------------------------------------------------------------------
__device__ __forceinline__ v8f wmma_tile(const float* __restrict__ A,
                                         const float* __restrict__ W,
                                         int m0, int n0, v8f c) {
    int lane = threadIdx.x & 31;
    int half = lane >> 4;           // 0 or 1
    int l    = lane & 15;
    const float* arow = A + (long)(m0 + l) * DIM;
    const float* wrow = W + (long)(n0 + l) * DIM;
    #pragma unroll
    for (int k = 0; k < DIM; k += 4) {
        int kk = k + (half << 1);
        v2f a = *(const v2f*)(arow + kk);
        v2f b = *(const v2f*)(wrow + kk);
        c = __builtin_amdgcn_wmma_f32_16x16x4_f32(
                /*neg_a=*/false, a, /*neg_b=*/false, b,
                /*c_mod=*/(short)0, c, /*reuse_a=*/false, /*reuse_b=*/false);
    }
    return c;
}

// ---------------------------------------------------------------------------
// Kernel 7: S = H@W1^T + X0@W2^T + b2 ; s = sigmoid(S) ; hb = x0 + s*(h - x0)
//   256 threads = 8 waves; block owns 16 rows; wave w owns column tile w*16.
// ---------------------------------------------------------------------------
__global__ void k_gate(const float* __restrict__ H, const float* __restrict__ X0,
                       const float* __restrict__ W1, const float* __restrict__ W2,
                       const float* __restrict__ b2, float* __restrict__ HB) {
    int m0 = blockIdx.x * 16;
    int n0 = (threadIdx.x >> 5) * 16;
    int lane = threadIdx.x & 31;
    int half = lane >> 4;
    int l    = lane & 15;

    v8f c = {};
    c = wmma_tile(H,  W1, m0, n0, c);
    c = wmma_tile(X0, W2, m0, n0, c);

    int col   = n0 + l;
    int rbase = m0 + (half << 3);
    float bias = b2[col];
    #pragma unroll
    for (int r = 0; r < 8; ++r) {
        int row = rbase + r;
        long idx = (long)row * DIM + col;
        float s = 1.0f / (1.0f + expf(-(c[r] + bias)));
        float h = H[idx];
        float x = X0[idx];
        HB[idx] = fmaf(s, h - x, x);   // h*s + x*(1-s)
    }
}

// ---------------------------------------------------------------------------
// Kernel 8: out = HB @ Weff^T   (Weff = 0.1*Wl + 0.9*I already folded)
// ---------------------------------------------------------------------------
__global__ void k_final(const float* __restrict__ HB, const float* __restrict__ Weff,
                        float* __restrict__ out) {
    int m0 = blockIdx.x * 16;
    int n0 = (threadIdx.x >> 5) * 16;
    int lane = threadIdx.x & 31;
    int half = lane >> 4;
    int l    = lane & 15;

    v8f c = {};
    c = wmma_tile(HB, Weff, m0, n0, c);

    int col   = n0 + l;
    int rbase = m0 + (half << 3);
    #pragma unroll
    for (int r = 0; r < 8; ++r) {
        out[(long)(rbase + r) * DIM + col] = c[r];
    }
}

// ---------------------------------------------------------------------------
// Host-side launcher
// ---------------------------------------------------------------------------
extern "C" void kernel_launch(void* const* d_in, const int* in_sizes, int n_in,
                              void* d_out, int out_size, void* d_ws, size_t ws_size,
                              hipStream_t stream) {
    const float* feat = (const float*)d_in[0];
    const float* x0   = (const float*)d_in[1];
    const int*   src  = (const int*)d_in[2];
    const int*   dst  = (const int*)d_in[3];
    const float* W1   = (const float*)d_in[4];
    const float* W2   = (const float*)d_in[5];
    const float* b2   = (const float*)d_in[6];
    const float* Wl   = (const float*)d_in[7];
    float* out = (float*)d_out;

    const long ND = (long)NNODES * DIM;
    float* ws   = (float*)d_ws;
    float* deg  = ws;                    // N floats (becomes norm)
    float* agg  = ws + NNODES;           // N*D floats (becomes H)
    float* hb   = agg + ND;              // N*D floats
    float* weff = hb + ND;               // 128*128 floats

    // 1. zero deg + agg  (contiguous: N + N*D floats, both /4 aligned)
    {
        long n4 = (NNODES + ND) / 4;
        int blocks = (int)((n4 + 255) / 256);
        if (blocks > 4096) blocks = 4096;           // grid-stride
        k_zero<<<blocks, 256, 0, stream>>>((float4*)deg, n4);
    }
    // 2. degree count
    k_degree<<<(NEDGES + 255) / 256, 256, 0, stream>>>(dst, deg, NEDGES);
    // 3. norm = rsqrt(clip(deg,1))
    k_norm<<<(NNODES + 255) / 256, 256, 0, stream>>>(deg, NNODES);
    // 4. W_eff
    k_weff<<<(DIM * DIM) / 256, 256, 0, stream>>>(Wl, weff);
    // 5. edge scatter (1 wave per edge, 8 edges per 256-thread block)
    k_scatter<<<(NEDGES + 7) / 8, 256, 0, stream>>>(feat, src, dst, deg, agg, NEDGES);
    // 6. H = agg * norm[dst-row]
    k_scale_rows<<<(int)(ND / 4 / 256), 256, 0, stream>>>((float4*)agg, deg);
    // 7. gated blend (two fused WMMA GEMMs + sigmoid)
    k_gate<<<NNODES / 16, 256, 0, stream>>>(agg, x0, W1, W2, b2, hb);
    // 8. final GEMM with folded residual
    k_final<<<NNODES / 16, 256, 0, stream>>>(hb, weff, out);
}